// SoftSOM_4114578669898
// MI455X (gfx1250) — compile-verified
//
#include <hip/hip_runtime.h>
#include <hip/hip_bf16.h>
#include <math.h>

typedef __attribute__((ext_vector_type(2))) float v2f;
typedef __attribute__((ext_vector_type(8))) float v8f;

#define B_ROWS 32768
#define NPROTO 256
#define DDIM   1024
#define EPS_TEMP 1e-3f
#define EPS_NORM 1e-8f

// ---------------------------------------------------------------------------
// Kernel 0: squared row norms. One wave (32 lanes) per row, 8 rows per block.
// ---------------------------------------------------------------------------
__global__ __launch_bounds__(256) void row_norms_kernel(
    const float* __restrict__ src, float* __restrict__ dst, int d) {
  const int wv   = threadIdx.x >> 5;
  const int lane = threadIdx.x & 31;
  const int row  = blockIdx.x * 8 + wv;
  const float* p = src + (size_t)row * d;
  float s = 0.f;
  for (int j = lane; j < d; j += 32) { float v = p[j]; s += v * v; }
  for (int off = 16; off; off >>= 1) s += __shfl_xor(s, off, 32);
  if (lane == 0) dst[row] = s;
}

// ---------------------------------------------------------------------------
// Kernel 1: fused dual-distance + softmax + gate + renormalize -> weights.
// Block = 256 threads (8 waves), 32 rows of x, 512 cols (256 P + 256 G).
// Double-buffered K loop unrolled x2 so the buffer index is a compile-time
// constant (LDS addresses = base + immediate). Each K-step loads all 8 B
// fragments before the 8 WMMAs so DS loads pipeline with partial dscnt waits.
// K-pair interleaved LDS ([kp][col][2]): fragment = one aligned ds_load_b64;
// +32-float stride pad puts the half-wave 128B runs on disjoint bank halves.
// ---------------------------------------------------------------------------
#define K1_ROWS 32
#define K1_KC   8
#define K1_KP   (K1_KC / 2)
#define XT_STRIDE (K1_ROWS * 2 + 32)   // 96 floats  (384 B: +128B bank shift)
#define BT_STRIDE (512 * 2 + 32)       // 1056 floats (4224 B: +128B bank shift)

union SmemK1 {
  struct {                                  // GEMM phase (~36.8 KB)
    float XtP[2][K1_KP][XT_STRIDE];
    float BtP[2][K1_KP][BT_STRIDE];
  } g;
  struct {                                  // epilogue phase (~35 KB)
    float dtot[K1_ROWS][NPROTO];
    float redm[K1_ROWS][8];
    float redA[K1_ROWS][8];
    float redB[K1_ROWS][8];
  } e;
};

template <int BUF>
__device__ __forceinline__ void k1_stage(SmemK1& sm, int k0,
                                         const float* __restrict__ x,
                                         const float* __restrict__ proto,
                                         const float* __restrict__ grid,
                                         int row0, int tid) {
  // XtP: 32 rows x 8 k -> 128 float2, threads 0..127.
  if (tid < 128) {
    int r  = tid >> 2;
    int kp = tid & 3;
    float2 v = *(const float2*)(x + (size_t)(row0 + r) * DDIM + k0 + kp * 2);
    *(float2*)&sm.g.XtP[BUF][kp][r * 2] = v;
  }
  // BtP: 512 cols x 8 k, 2 columns per thread via float4 loads.
  #pragma unroll
  for (int cc = 0; cc < 2; cc++) {
    int c = tid + cc * 256;
    const float* src = (c < 256) ? proto + (size_t)c * DDIM + k0
                                 : grid + (size_t)(c - 256) * DDIM + k0;
    float4 v0 = *(const float4*)(src);
    float4 v1 = *(const float4*)(src + 4);
    *(float2*)&sm.g.BtP[BUF][0][c * 2] = make_float2(v0.x, v0.y);
    *(float2*)&sm.g.BtP[BUF][1][c * 2] = make_float2(v0.z, v0.w);
    *(float2*)&sm.g.BtP[BUF][2][c * 2] = make_float2(v1.x, v1.y);
    *(float2*)&sm.g.BtP[BUF][3][c * 2] = make_float2(v1.z, v1.w);
  }
}

template <int BUF>
__device__ __forceinline__ void k1_compute(const SmemK1& sm, v8f* acc,
                                           int rt, int cseg, int lm, int lh) {
  #pragma unroll
  for (int ks = 0; ks < K1_KC; ks += 4) {
    const int kp = (ks >> 1) + lh;
    v2f a = *(const v2f*)&sm.g.XtP[BUF][kp][(rt * 16 + lm) * 2];
    v2f b[8];
    #pragma unroll
    for (int ti = 0; ti < 8; ti++) {
      int c = cseg * 128 + ti * 16 + lm;
      b[ti] = *(const v2f*)&sm.g.BtP[BUF][kp][c * 2];
    }
    #pragma unroll
    for (int ti = 0; ti < 8; ti++) {
      acc[ti] = __builtin_amdgcn_wmma_f32_16x16x4_f32(
          false, a, false, b[ti], (short)0, acc[ti], false, false);
    }
  }
}

__global__ __launch_bounds__(256) void dist_softmax_kernel(
    const float* __restrict__ x, const float* __restrict__ proto,
    const float* __restrict__ grid, const float* __restrict__ temp_raw,
    const float* __restrict__ gate_logits, const float* __restrict__ x2,
    const float* __restrict__ p2, const float* __restrict__ g2,
    float* __restrict__ wout) {
  __shared__ SmemK1 sm;

  const int tid  = threadIdx.x;
  const int wv   = tid >> 5;
  const int lane = tid & 31;
  const int row0 = blockIdx.x * K1_ROWS;

  const int rt   = wv & 1;     // row tile within block (0/1)
  const int cseg = wv >> 1;    // 128-column segment (0..3) of 512
  const int lm   = lane & 15;
  const int lh   = lane >> 4;  // half-wave select
  const int rhi  = lh ? 8 : 0;

  v8f acc[8] = {};             // 8 16x16 f32 accumulator tiles per wave

  // Prologue: stage chunk 0 into buffer 0.
  k1_stage<0>(sm, 0, x, proto, grid, row0, tid);
  __syncthreads();

  // 128 chunks, 2 per iteration; buffer index is a literal in each phase.
  for (int k0 = 0; k0 < DDIM; k0 += 2 * K1_KC) {
    if (k0 + K1_KC < DDIM) {
      __builtin_prefetch(x + (size_t)(row0 + (tid >> 2)) * DDIM
                             + k0 + 2 * K1_KC, 0, 0);
      __builtin_prefetch((tid < 128 ? proto : grid)
                             + (size_t)(tid & 255) * DDIM + k0 + 2 * K1_KC, 0, 0);
      k1_stage<1>(sm, k0 + K1_KC, x, proto, grid, row0, tid);
    }
    k1_compute<0>(sm, acc, rt, cseg, lm, lh);
    __syncthreads();
    if (k0 + 2 * K1_KC < DDIM) {
      k1_stage<0>(sm, k0 + 2 * K1_KC, x, proto, grid, row0, tid);
    }
    k1_compute<1>(sm, acc, rt, cseg, lm, lh);
    __syncthreads();
  }

  // Epilogue: d = sqrt(max(|x|^2 + |n|^2 - 2ab, 0)), overwrite acc with d.
  float x2v[8];
  {
    int rbase = row0 + rt * 16 + rhi;
    #pragma unroll
    for (int j = 0; j < 8; j++) x2v[j] = x2[rbase + j];
  }
  const bool isG = (cseg >= 2);
  #pragma unroll
  for (int ti = 0; ti < 8; ti++) {
    int c = cseg * 128 + ti * 16 + lm;     // 0..511
    int n = c & 255;
    float n2 = isG ? g2[n] : p2[n];
    #pragma unroll
    for (int j = 0; j < 8; j++) {
      float d2 = x2v[j] + n2 - 2.0f * acc[ti][j];
      acc[ti][j] = sqrtf(fmaxf(d2, 0.0f));
    }
  }
  // P-half waves store, then G-half waves accumulate (disjoint per phase).
  // (GEMM staging buffers are dead; sm.e aliases them — barriers separate.)
  if (!isG) {
    #pragma unroll
    for (int ti = 0; ti < 8; ti++) {
      int n  = cseg * 128 + ti * 16 + lm;
      int rb = rt * 16 + rhi;
      #pragma unroll
      for (int j = 0; j < 8; j++) sm.e.dtot[rb + j][n] = acc[ti][j];
    }
  }
  __syncthreads();
  if (isG) {
    #pragma unroll
    for (int ti = 0; ti < 8; ti++) {
      int n  = (cseg - 2) * 128 + ti * 16 + lm;
      int rb = rt * 16 + rhi;
      #pragma unroll
      for (int j = 0; j < 8; j++) sm.e.dtot[rb + j][n] += acc[ti][j];
    }
  }
  __syncthreads();

  // Softmax(-d/T)*gate, renormalized. 8 threads per row, 32 cols each.
  float T;
  {
    float tr = temp_raw[0];
    float sg = 1.0f / (1.0f + __expf(-tr));
    T = sg * (1.0f - EPS_TEMP) + EPS_TEMP;
  }
  const float invT = 1.0f / T;
  const int r   = tid >> 3;
  const int seg = tid & 7;

  float dmin = 3.0e38f;
  for (int jc = 0; jc < 32; jc++)
    dmin = fminf(dmin, sm.e.dtot[r][seg * 32 + jc]);
  sm.e.redm[r][seg] = dmin;
  __syncthreads();
  float rowmin = sm.e.redm[r][0];
  #pragma unroll
  for (int q = 1; q < 8; q++) rowmin = fminf(rowmin, sm.e.redm[r][q]);

  float s1 = 0.f, s2 = 0.f;
  for (int jc = 0; jc < 32; jc++) {
    int n = seg * 32 + jc;
    float e = __expf((rowmin - sm.e.dtot[r][n]) * invT);
    float g = 1.0f / (1.0f + __expf(-gate_logits[n]));
    s1 += e;
    s2 += e * g;
  }
  sm.e.redA[r][seg] = s1;
  sm.e.redB[r][seg] = s2;
  __syncthreads();
  float S1 = 0.f, S2 = 0.f;
  #pragma unroll
  for (int q = 0; q < 8; q++) { S1 += sm.e.redA[r][q]; S2 += sm.e.redB[r][q]; }
  // weights = (e*g/S1) / (S2/S1 + eps) = e*g / (S2 + eps*S1)
  const float dinv = 1.0f / (S2 + EPS_NORM * S1);

  float* wrow = wout + (size_t)(row0 + r) * NPROTO;
  for (int jc = 0; jc < 32; jc++) {
    int n = seg * 32 + jc;
    float e = __expf((rowmin - sm.e.dtot[r][n]) * invT);
    float g = 1.0f / (1.0f + __expf(-gate_logits[n]));
    wrow[n] = e * g * dinv;
  }
}

// ---------------------------------------------------------------------------
// Kernel 2: blended = weights[32768,256] @ proto[256,1024] via f32 WMMA.
// Block = 256 threads, 32-row x 256-col output tile, K=256 in 16 chunks,
// double-buffered with literal buffer indices (loop unrolled x2).
// ---------------------------------------------------------------------------
#define K2_ROWS 32
#define K2_COLS 256
#define K2_KC   16
#define K2_KP   (K2_KC / 2)
#define WL_STRIDE (K2_ROWS * 2 + 32)    // 96 floats
#define PT_STRIDE (K2_COLS * 2 + 32)    // 544 floats

struct SmemK2 {
  float WlP[2][K2_KP][WL_STRIDE];   // ~6 KB
  float PtP[2][K2_KP][PT_STRIDE];   // ~35 KB
};

template <int BUF>
__device__ __forceinline__ void k2_stage(SmemK2& sm, int k0,
                                         const float* __restrict__ w,
                                         const float* __restrict__ proto,
                                         int row0, int col0, int tid) {
  // WlP: 16k x 32r, one float2 per thread, K-pair interleaved.
  {
    int r  = tid >> 3;
    int kk = (tid & 7) * 2;
    float2 v = *(const float2*)(w + (size_t)(row0 + r) * NPROTO + k0 + kk);
    *(float2*)&sm.WlP[BUF][kk >> 1][r * 2] = v;
  }
  // PtP: per k-pair, 256 threads read two coalesced 256-float rows.
  #pragma unroll
  for (int p = 0; p < K2_KP; p++) {
    float a0 = proto[(size_t)(k0 + 2 * p) * DDIM + col0 + tid];
    float a1 = proto[(size_t)(k0 + 2 * p + 1) * DDIM + col0 + tid];
    *(float2*)&sm.PtP[BUF][p][tid * 2] = make_float2(a0, a1);
  }
}

template <int BUF>
__device__ __forceinline__ void k2_compute(const SmemK2& sm, v8f* acc,
                                           int rt, int cq, int lm, int lh) {
  #pragma unroll
  for (int ks = 0; ks < K2_KC; ks += 4) {
    const int kp = (ks >> 1) + lh;
    v2f a = *(const v2f*)&sm.WlP[BUF][kp][(rt * 16 + lm) * 2];
    v2f b[4];
    #pragma unroll
    for (int ti = 0; ti < 4; ti++) {
      int c = cq * 64 + ti * 16 + lm;
      b[ti] = *(const v2f*)&sm.PtP[BUF][kp][c * 2];
    }
    #pragma unroll
    for (int ti = 0; ti < 4; ti++) {
      acc[ti] = __builtin_amdgcn_wmma_f32_16x16x4_f32(
          false, a, false, b[ti], (short)0, acc[ti], false, false);
    }
  }
}

__global__ __launch_bounds__(256) void blend_kernel(
    const float* __restrict__ w, const float* __restrict__ proto,
    float* __restrict__ out) {
  __shared__ SmemK2 sm;

  const int tid  = threadIdx.x;
  const int wv   = tid >> 5;
  const int lane = tid & 31;
  const int row0 = blockIdx.y * K2_ROWS;
  const int col0 = blockIdx.x * K2_COLS;

  const int rt  = wv & 1;
  const int cq  = wv >> 1;            // 0..3, four 16-col tiles each
  const int lm  = lane & 15;
  const int lh  = lane >> 4;
  const int rhi = lh ? 8 : 0;

  v8f acc[4] = {};

  k2_stage<0>(sm, 0, w, proto, row0, col0, tid);
  __syncthreads();

  for (int k0 = 0; k0 < NPROTO; k0 += 2 * K2_KC) {
    if (k0 + K2_KC < NPROTO) {
      __builtin_prefetch(w + (size_t)(row0 + (tid >> 3)) * NPROTO
                             + k0 + 2 * K2_KC, 0, 0);
      __builtin_prefetch(proto + (size_t)(k0 + 2 * K2_KC + (tid >> 4)) * DDIM
                             + col0 + (tid & 15) * 16, 0, 0);
      k2_stage<1>(sm, k0 + K2_KC, w, proto, row0, col0, tid);
    }
    k2_compute<0>(sm, acc, rt, cq, lm, lh);
    __syncthreads();
    if (k0 + 2 * K2_KC < NPROTO) {
      k2_stage<0>(sm, k0 + 2 * K2_KC, w, proto, row0, col0, tid);
    }
    k2_compute<1>(sm, acc, rt, cq, lm, lh);
    __syncthreads();
  }

  const int rb = row0 + rt * 16 + rhi;
  #pragma unroll
  for (int ti = 0; ti < 4; ti++) {
    int c = col0 + cq * 64 + ti * 16 + lm;
    #pragma unroll
    for (int j = 0; j < 8; j++) {
      out[(size_t)(rb + j) * DDIM + c] = acc[ti][j];
    }
  }
}

// ---------------------------------------------------------------------------
extern "C" void kernel_launch(void* const* d_in, const int* in_sizes, int n_in,
                              void* d_out, int out_size, void* d_ws, size_t ws_size,
                              hipStream_t stream) {
  const float* x           = (const float*)d_in[0];
  const float* proto       = (const float*)d_in[1];
  const float* grid        = (const float*)d_in[2];
  const float* temp_raw    = (const float*)d_in[3];
  const float* gate_logits = (const float*)d_in[4];

  float* out      = (float*)d_out;
  float* blended  = out;                              // [B, D]
  float* weights  = out + (size_t)B_ROWS * DDIM;      // [B, N]

  float* ws = (float*)d_ws;
  float* x2 = ws;                  // [B]
  float* p2 = ws + B_ROWS;         // [N]
  float* g2 = p2 + NPROTO;         // [N]

  row_norms_kernel<<<B_ROWS / 8, 256, 0, stream>>>(x, x2, DDIM);
  row_norms_kernel<<<NPROTO / 8, 256, 0, stream>>>(proto, p2, DDIM);
  row_norms_kernel<<<NPROTO / 8, 256, 0, stream>>>(grid, g2, DDIM);

  dist_softmax_kernel<<<B_ROWS / K1_ROWS, 256, 0, stream>>>(
      x, proto, grid, temp_raw, gate_logits, x2, p2, g2, weights);

  blend_kernel<<<dim3(DDIM / K2_COLS, B_ROWS / K2_ROWS), 256, 0, stream>>>(
      weights, proto, blended);
}